// DetLoss_4578435138206
// MI455X (gfx1250) — compile-verified
//
#include <hip/hip_runtime.h>
#include <math.h>

// Problem shape constants (fixed by the reference harness).
#define KB 8    // batch
#define KN 16   // annotations per image
#define KC 4    // classes

typedef float v8f __attribute__((ext_vector_type(8)));
typedef float v2f __attribute__((ext_vector_type(2)));

// ---------------------------------------------------------------------------
// Wave-level f32 sum using V_WMMA_F32_16X16X4_F32.
// A = all-ones 16x4 (layout-independent), B holds one lane value per lane
// (B VGPR0 = x, VGPR1 = 0). D[m][n] = sum over k of B[k][n] -> the 16 column
// sums partition the 32 lane values exactly, independent of the B slot
// layout. Row m=0 of D (d[0] on lanes 0..15, mirrored on 16..31) holds all
// 16 column sums, so 4 xor-shuffles finish the exact f32 reduction.
// Requires EXEC == all ones at the call site (call uniformly!).
// ---------------------------------------------------------------------------
__device__ __forceinline__ float wave_sum(float x) {
  v2f a; a.x = 1.0f; a.y = 1.0f;
  v2f b; b.x = x;    b.y = 0.0f;
  v8f c = {};
  v8f d = __builtin_amdgcn_wmma_f32_16x16x4_f32(
      /*neg_a=*/false, a, /*neg_b=*/false, b,
      /*c_mod=*/(short)0, c, /*reuse_a=*/false, /*reuse_b=*/false);
  float v = d[0];
  v += __shfl_xor(v, 8, 32);
  v += __shfl_xor(v, 4, 32);
  v += __shfl_xor(v, 2, 32);
  v += __shfl_xor(v, 1, 32);
  return v;  // full-wave sum, replicated in every lane
}

__device__ __forceinline__ float block_sum(float x, float* s_red) {
  float v = wave_sum(x);
  const int wave = threadIdx.x >> 5;
  const int lane = threadIdx.x & 31;
  if (lane == 0) s_red[wave] = v;
  __syncthreads();
  float r = 0.0f;
  if (wave == 0) {  // uniform within wave 0 -> EXEC all ones for WMMA
    float t = (lane < (int)(blockDim.x >> 5)) ? s_red[lane] : 0.0f;
    r = wave_sum(t);
  }
  __syncthreads();  // allow s_red reuse by the next call
  return r;         // valid on wave 0 (thread 0 consumes it)
}

// ---------------------------------------------------------------------------
// Workspace init (harness poisons d_ws once; we must zero every launch).
// ---------------------------------------------------------------------------
__global__ void k_init(unsigned long long* __restrict__ annKey,
                       float* __restrict__ accum) {
  const int i = threadIdx.x;
  if (i < KB * KN) annKey[i] = 0ull;
  if (i < KB * 3)  accum[i]  = 0.0f;
}

// ---------------------------------------------------------------------------
// Pass 1: per-(image, annotation) max IoU over ALL anchors, with argmax and
// first-index tie-break. Key = (float_bits(iou) << 32) | ~anchor_idx.
// IoU >= 0 so float bits are monotone as u32; ~idx makes the smaller index
// win ties under unsigned max. Reduced via LDS u64 atomicMax then one global
// u64 atomicMax per (image, annotation).
// ---------------------------------------------------------------------------
__global__ void k_annmax(const float* __restrict__ anchors,
                         const float* __restrict__ ann,
                         unsigned long long* __restrict__ annKey,
                         int A) {
  __shared__ float s_bb[KN][5];   // x1,y1,x2,y2,area
  __shared__ int   s_val[KN];
  __shared__ unsigned long long s_key[KN];

  const int b   = blockIdx.y;
  const int tid = threadIdx.x;

  if (tid < KN) {
    const float* r = ann + ((size_t)b * KN + tid) * 6;
    float ax = r[0], ay = r[1], th = r[2], ln = r[3], cf = r[4];
    float dx = fabsf(0.5f * ln * cosf(th));
    float dy = fabsf(0.5f * ln * sinf(th));
    s_bb[tid][0] = ax - dx; s_bb[tid][1] = ay - dy;
    s_bb[tid][2] = ax + dx; s_bb[tid][3] = ay + dy;
    s_bb[tid][4] = (2.0f * dx) * (2.0f * dy);
    s_val[tid]   = (cf != -1.0f) ? 1 : 0;
    s_key[tid]   = 0ull;
  }
  __syncthreads();

  unsigned long long best[KN];
#pragma unroll
  for (int n = 0; n < KN; ++n) best[n] = 0ull;

  for (int a = blockIdx.x * blockDim.x + tid; a < A;
       a += gridDim.x * blockDim.x) {
    const float4 ap = ((const float4*)anchors)[a];
    const float aarea = (ap.z - ap.x) * (ap.w - ap.y);
    const unsigned long long ktag = (unsigned long long)(~(unsigned)a);
#pragma unroll
    for (int n = 0; n < KN; ++n) {
      if (!s_val[n]) continue;  // masked IoU = -1 can never win vs key>=0
      float iw = fminf(ap.z, s_bb[n][2]) - fmaxf(ap.x, s_bb[n][0]);
      float ih = fminf(ap.w, s_bb[n][3]) - fmaxf(ap.y, s_bb[n][1]);
      iw = fmaxf(iw, 0.0f); ih = fmaxf(ih, 0.0f);
      float inter = iw * ih;
      float iou = inter / fmaxf(aarea + s_bb[n][4] - inter, 1e-8f);
      unsigned long long key =
          ((unsigned long long)__float_as_uint(iou) << 32) | ktag;
      best[n] = (key > best[n]) ? key : best[n];
    }
  }

#pragma unroll
  for (int n = 0; n < KN; ++n) atomicMax(&s_key[n], best[n]);
  __syncthreads();
  if (tid < KN) atomicMax(&annKey[b * KN + tid], s_key[tid]);
}

// ---------------------------------------------------------------------------
// Pass 2: per-anchor focal + smoothL1, all 8 images per anchor so anchors
// are read once. Regressions are only loaded for positive anchors
// (predicated) — most of that 32 MB is never fetched.
// ---------------------------------------------------------------------------
struct AnnInfo {
  float bx1, by1, bx2, by2, barea;  // derived bbox
  float ax, ay, ath, alen;          // raw annotation (regression target)
  int   cls;                        // -1 if invalid
  int   forcedIdx;                  // forced anchor index, or -1
};

__global__ void k_loss(const float* __restrict__ cls_,
                       const float* __restrict__ reg_,
                       const float* __restrict__ anchors,
                       const float* __restrict__ ann,
                       const unsigned long long* __restrict__ annKey,
                       float* __restrict__ accum,
                       int A) {
  __shared__ AnnInfo s_ann[KB * KN];
  __shared__ float   s_red[32];

  const int tid = threadIdx.x;
  if (tid < KB * KN) {
    const float* r = ann + (size_t)tid * 6;
    AnnInfo ai;
    ai.ax = r[0]; ai.ay = r[1]; ai.ath = r[2]; ai.alen = r[3];
    float cf = r[4];
    bool valid = (cf != -1.0f);
    ai.cls = valid ? (int)cf : -1;
    float dx = fabsf(0.5f * ai.alen * cosf(ai.ath));
    float dy = fabsf(0.5f * ai.alen * sinf(ai.ath));
    ai.bx1 = ai.ax - dx; ai.by1 = ai.ay - dy;
    ai.bx2 = ai.ax + dx; ai.by2 = ai.ay + dy;
    ai.barea = (2.0f * dx) * (2.0f * dy);
    unsigned long long key = annKey[tid];
    float mx = __uint_as_float((unsigned)(key >> 32));
    ai.forcedIdx = (valid && mx < 0.5f)
                       ? (int)(~(unsigned)(key & 0xffffffffu)) : -1;
    s_ann[tid] = ai;
  }
  __syncthreads();

  const int  a      = blockIdx.x * blockDim.x + tid;
  const bool active = (a < A);

  // Anchor geometry (reference coordinates_transform), computed once.
  float4 ap = make_float4(0.f, 0.f, 1.f, 1.f);
  if (active) ap = ((const float4*)anchors)[a];
  const float aarea = (ap.z - ap.x) * (ap.w - ap.y);
  const float w = ap.z - ap.x, h = ap.w - ap.y;
  const float ctrx = 0.5f * (ap.x + ap.z), ctry = 0.5f * (ap.y + ap.w);
  const float L = sqrtf(w * w + h * h);
  float th;
  if (w == 0.0f)
    th = (h > 0.0f) ? 1.57079632679f : (h < 0.0f ? -1.57079632679f : 0.0f);
  else
    th = atanf(h / w);

  for (int b = 0; b < KB; ++b) {
    // Prefetch next image's classification row (global_prefetch_b8).
    if (active && b + 1 < KB)
      __builtin_prefetch(cls_ + ((size_t)(b + 1) * A + a) * KC, 0, 1);

    const AnnInfo* sb = &s_ann[b * KN];
    float clsSum = 0.0f, regSum = 0.0f, nPos = 0.0f;

    if (active) {
      // IoU max / argmax (first index wins ties, masked invalid -> -1).
      float m = -3.0e38f; int arg = 0;
#pragma unroll
      for (int n = 0; n < KN; ++n) {
        float v;
        if (sb[n].cls < 0) {
          v = -1.0f;
        } else {
          float iw = fminf(ap.z, sb[n].bx2) - fmaxf(ap.x, sb[n].bx1);
          float ih = fminf(ap.w, sb[n].by2) - fmaxf(ap.y, sb[n].by1);
          iw = fmaxf(iw, 0.0f); ih = fmaxf(ih, 0.0f);
          float inter = iw * ih;
          v = inter / fmaxf(aarea + sb[n].barea - inter, 1e-8f);
        }
        if (v > m) { m = v; arg = n; }
      }

      // Forced assignment: every annotation whose global argmax is this
      // anchor contributes a one-hot class (matches scatter semantics);
      // the last one supplies the regression target.
      int fmask = 0, lastN = -1;
#pragma unroll
      for (int n = 0; n < KN; ++n)
        if (sb[n].forcedIdx == a) { fmask |= 1 << sb[n].cls; lastN = n; }

      const bool pos = (m >= 0.5f), neg = (m < 0.4f);
      int  tcls[KC];
      bool positive;
      int  assignN;
      if (lastN >= 0) {
#pragma unroll
        for (int c = 0; c < KC; ++c) tcls[c] = (fmask >> c) & 1;
        positive = true; assignN = lastN;
      } else if (pos) {
        const int ac = sb[arg].cls;
#pragma unroll
        for (int c = 0; c < KC; ++c) tcls[c] = (c == ac) ? 1 : 0;
        positive = true; assignN = arg;
      } else {
#pragma unroll
        for (int c = 0; c < KC; ++c) tcls[c] = neg ? 0 : -1;
        positive = false; assignN = arg;
      }

      // Focal loss (ALPHA=0.25, GAMMA=2).
      const float4 cv = *(const float4*)(cls_ + ((size_t)b * A + a) * KC);
      const float carr[KC] = {cv.x, cv.y, cv.z, cv.w};
#pragma unroll
      for (int c = 0; c < KC; ++c) {
        if (tcls[c] == -1) continue;
        float p = fminf(fmaxf(carr[c], 1e-4f), 1.0f - 1e-4f);
        if (tcls[c] == 1)
          clsSum += 0.25f * (1.0f - p) * (1.0f - p) * (-logf(p));
        else
          clsSum += 0.75f * p * p * (-logf(1.0f - p));
      }

      // Regression smooth-L1, only for positives (predicated load).
      if (positive) {
        nPos = 1.0f;
        const float4 rv = *(const float4*)(reg_ + ((size_t)b * A + a) * 4);
        const AnnInfo& an = sb[assignN];
        const float prd[4] = {rv.x * w + ctrx, rv.y * h + ctry,
                              rv.z + th, expf(rv.w) * L};
        const float tgt[4] = {an.ax, an.ay, an.ath, an.alen};
        float s = 0.0f;
#pragma unroll
        for (int k = 0; k < 4; ++k) {
          float d = prd[k] - tgt[k];
          float ad = fabsf(d);
          s += (ad < 1.0f) ? 0.5f * d * d : ad - 0.5f;
        }
        regSum = 0.25f * s;
      }
    }

    // Uniform calls: WMMA-based block reductions, then one atomic each.
    const float rc = block_sum(clsSum, s_red);
    const float rr = block_sum(regSum, s_red);
    const float rn = block_sum(nPos,   s_red);
    if (tid == 0) {
      atomicAdd(&accum[b * 3 + 0], rc);
      atomicAdd(&accum[b * 3 + 1], rr);
      atomicAdd(&accum[b * 3 + 2], rn);
    }
  }
}

// ---------------------------------------------------------------------------
// Finalize: per-image normalization + batch means.
// ---------------------------------------------------------------------------
__global__ void k_final(const float* __restrict__ accum,
                        float* __restrict__ out) {
  if (threadIdx.x == 0) {
    float cs = 0.0f, rs = 0.0f;
    for (int b = 0; b < KB; ++b) {
      float np = fmaxf(accum[b * 3 + 2], 1.0f);
      cs += accum[b * 3 + 0] / np;
      rs += accum[b * 3 + 1] / np;
    }
    out[0] = cs / (float)KB;
    out[1] = rs / (float)KB;
  }
}

// ---------------------------------------------------------------------------
extern "C" void kernel_launch(void* const* d_in, const int* in_sizes, int n_in,
                              void* d_out, int out_size, void* d_ws,
                              size_t ws_size, hipStream_t stream) {
  const float* cls_    = (const float*)d_in[0];  // (B, A, C) f32
  const float* reg_    = (const float*)d_in[1];  // (B, A, 4) f32
  const float* anchors = (const float*)d_in[2];  // (A, 4)    f32
  const float* ann     = (const float*)d_in[3];  // (B, N, 6) f32
  const int A = in_sizes[2] / 4;

  unsigned long long* annKey = (unsigned long long*)d_ws;
  float* accum = (float*)((char*)d_ws + KB * KN * sizeof(unsigned long long));
  float* out = (float*)d_out;

  const int blk = 256;                // 8 wave32 waves per workgroup
  const int nb  = (A + blk - 1) / blk;

  hipLaunchKernelGGL(k_init, dim3(1), dim3(128), 0, stream, annKey, accum);
  hipLaunchKernelGGL(k_annmax, dim3(nb, KB), dim3(blk), 0, stream,
                     anchors, ann, annKey, A);
  hipLaunchKernelGGL(k_loss, dim3(nb), dim3(blk), 0, stream,
                     cls_, reg_, anchors, ann, annKey, accum, A);
  hipLaunchKernelGGL(k_final, dim3(1), dim3(32), 0, stream, accum, out);
}